// DrBCRNN_63934883168896
// MI455X (gfx1250) — compile-verified
//
#include <hip/hip_runtime.h>

typedef __attribute__((ext_vector_type(2))) float v2f;
typedef __attribute__((ext_vector_type(8))) float v8f;

#define UNITS 128
#define REPS 5
#define LDSW 132  // padded LDS row stride (floats) -> conflict-free column reads

// ---------------- zero pool ----------------
__global__ __launch_bounds__(256) void zero_f4(float4* __restrict__ p, int n4) {
  int i = blockIdx.x * blockDim.x + threadIdx.x;
  if (i < n4) p[i] = make_float4(0.f, 0.f, 0.f, 0.f);
}

// ------------- edge gather/scatter -------------
// one wave32 per edge; each lane moves one float4 (32*4 = 128 floats/row)
__global__ __launch_bounds__(256) void edge_scatter(
    const float* __restrict__ h, const int* __restrict__ src,
    const int* __restrict__ dst, const float* __restrict__ ew,
    float* __restrict__ pool, int E) {
  int e = blockIdx.x * 8 + (threadIdx.x >> 5);
  if (e >= E) return;
  int lane = threadIdx.x & 31;
  int s = src[e];
  int d = dst[e];
  float w = ew[e];
  float4 v = *((const float4*)(h + (size_t)s * UNITS) + lane);
  float* pp = pool + (size_t)d * UNITS + lane * 4;
  unsafeAtomicAdd(pp + 0, v.x * w);
  unsafeAtomicAdd(pp + 1, v.y * w);
  unsafeAtomicAdd(pp + 2, v.z * w);
  unsafeAtomicAdd(pp + 3, v.w * w);
}

__device__ __forceinline__ float sigmoidf_(float x) {
  return 1.f / (1.f + __expf(-x));
}

// ------------- fused GEMM1 + GEMM2 + GRU + l2norm -------------
// 4 waves/block, each wave owns one 16-row tile.
// WMMA f32 16x16x4: A frag lane l<16 -> A[M=l, K=k0+{0,1}], lane>=16 -> K=k0+2+{0,1}
//                   B frag vgpr0: B[K=k0+2*half, N=n0+l15], vgpr1: next K row
//                   C/D: vgpr v -> M = v + 8*half, N = n0 + l15
__global__ __launch_bounds__(128) void gemm_gru(
    const float* __restrict__ pool, const float* __restrict__ Wn,
    const float* __restrict__ bn, const float* __restrict__ Wg,
    const float* __restrict__ bin, const float* __restrict__ brec,
    float* __restrict__ hout, float* __restrict__ out, int rep, int nrows) {
  __shared__ float lds[4][16 * LDSW];
  const int wave = threadIdx.x >> 5;
  const int lane = threadIdx.x & 31;
  const int half = lane >> 4;
  const int l15 = lane & 15;

  int tile = blockIdx.x * 4 + wave;
  int lastTile = (nrows >> 4) - 1;
  if (tile > lastTile) tile = lastTile;  // clamp: duplicate work, identical writes
  const int rowBase = tile * 16;
  float* slab = lds[wave];

  // Phase 0: stage pool tile (16 x 128) into LDS (row stride LDSW)
  {
    const float4* src4 = (const float4*)(pool + (size_t)rowBase * UNITS);
#pragma unroll
    for (int i = 0; i < 16; ++i) {
      int idx = i * 32 + lane;  // 0..511 float4 slots
      int r = idx >> 5;
      int c = idx & 31;
      *(float4*)(slab + r * LDSW + c * 4) = src4[idx];
    }
  }

  // Phase 1: node = pool @ W_node + b_node   (8 column tiles of 16)
  float node[8][8];
#pragma unroll
  for (int ct = 0; ct < 8; ++ct) {
    v8f acc = {};
    const float* ap = slab + l15 * LDSW + 2 * half;
    const float* bp = Wn + 16 * ct + l15 + (size_t)(2 * half) * UNITS;
    for (int kk = 0; kk < 32; ++kk) {
      v2f a = *(const v2f*)(ap + 4 * kk);
      v2f b;
      b[0] = bp[(size_t)(4 * kk) * UNITS];
      b[1] = bp[(size_t)(4 * kk + 1) * UNITS];
      acc = __builtin_amdgcn_wmma_f32_16x16x4_f32(false, a, false, b, (short)0,
                                                  acc, false, false);
    }
    float bc = bn[16 * ct + l15];
#pragma unroll
    for (int v = 0; v < 8; ++v) node[ct][v] = acc[v] + bc;
  }

  // Swap node tile into the LDS slab (pool tile is dead now)
#pragma unroll
  for (int ct = 0; ct < 8; ++ct) {
#pragma unroll
    for (int v = 0; v < 8; ++v)
      slab[(v + 8 * half) * LDSW + 16 * ct + l15] = node[ct][v];
  }

  // Phase 2: gx = node @ W_gru (+biases) -> GRU nonlinearity -> comb
  float comb[8][8];
  float ssq[8];
#pragma unroll
  for (int v = 0; v < 8; ++v) ssq[v] = 0.f;

#pragma unroll
  for (int ct = 0; ct < 8; ++ct) {
    v8f az = {}, ar = {}, ah = {};
    const float* ap = slab + l15 * LDSW + 2 * half;
    const float* bp = Wg + 16 * ct + l15 + (size_t)(2 * half) * (3 * UNITS);
    for (int kk = 0; kk < 32; ++kk) {
      v2f a = *(const v2f*)(ap + 4 * kk);
      const float* b0 = bp + (size_t)(4 * kk) * (3 * UNITS);
      const float* b1 = bp + (size_t)(4 * kk + 1) * (3 * UNITS);
      v2f bz; bz[0] = b0[0];         bz[1] = b1[0];
      v2f br; br[0] = b0[UNITS];     br[1] = b1[UNITS];
      v2f bh; bh[0] = b0[2 * UNITS]; bh[1] = b1[2 * UNITS];
      az = __builtin_amdgcn_wmma_f32_16x16x4_f32(false, a, false, bz, (short)0,
                                                 az, false, false);
      ar = __builtin_amdgcn_wmma_f32_16x16x4_f32(false, a, false, br, (short)0,
                                                 ar, false, false);
      ah = __builtin_amdgcn_wmma_f32_16x16x4_f32(false, a, false, bh, (short)0,
                                                 ah, false, false);
    }
    int col = 16 * ct + l15;
    float bz_ = bin[col] + brec[col];
    float br_ = bin[UNITS + col] + brec[UNITS + col];
    float bh_in = bin[2 * UNITS + col];
    float bh_rec = brec[2 * UNITS + col];
#pragma unroll
    for (int v = 0; v < 8; ++v) {
      float z = sigmoidf_(az[v] + bz_);
      float r = sigmoidf_(ar[v] + br_);
      float hh = tanhf(ah[v] + bh_in + r * bh_rec);
      float c = (1.f - z) * hh;
      comb[ct][v] = c;
      ssq[v] += c * c;
    }
  }

  // Row sum-of-squares: reduce across the 16 lanes of each half-wave
#pragma unroll
  for (int v = 0; v < 8; ++v) {
    float s = ssq[v];
    s += __shfl_xor(s, 1, 32);
    s += __shfl_xor(s, 2, 32);
    s += __shfl_xor(s, 4, 32);
    s += __shfl_xor(s, 8, 32);
    ssq[v] = s;
  }

  // Normalize + write h (contiguous) and out slice [n, rep*128 + col]
#pragma unroll
  for (int v = 0; v < 8; ++v) {
    float inv = rsqrtf(fmaxf(ssq[v], 1e-12f));
    int row = rowBase + v + 8 * half;
#pragma unroll
    for (int ct = 0; ct < 8; ++ct) {
      int col = 16 * ct + l15;
      float hval = comb[ct][v] * inv;
      hout[(size_t)row * UNITS + col] = hval;
      out[(size_t)row * (UNITS * REPS) + rep * UNITS + col] = hval;
    }
  }
}

// ---------------- driver ----------------
extern "C" void kernel_launch(void* const* d_in, const int* in_sizes, int n_in,
                              void* d_out, int out_size, void* d_ws,
                              size_t ws_size, hipStream_t stream) {
  const float* message = (const float*)d_in[0];
  const int* src = (const int*)d_in[1];
  const int* dst = (const int*)d_in[2];
  const float* ew = (const float*)d_in[3];
  const float* Wn = (const float*)d_in[4];
  const float* bn = (const float*)d_in[5];
  const float* Wg = (const float*)d_in[6];
  // d_in[7] = U_gru: unused (h_prev == 0 in the reference)
  const float* bin = (const float*)d_in[8];
  const float* brec = (const float*)d_in[9];
  float* out = (float*)d_out;

  const int N = in_sizes[0] / UNITS;
  const int E = in_sizes[1];

  float* pool = (float*)d_ws;                  // N*128 floats
  float* hbuf = pool + (size_t)N * UNITS;      // N*128 floats

  const int n4 = N * UNITS / 4;
  const int zeroBlocks = (n4 + 255) / 256;
  const int edgeBlocks = (E + 7) / 8;
  const int tiles = (N + 15) / 16;
  const int gemmBlocks = (tiles + 3) / 4;

  for (int rep = 0; rep < REPS; ++rep) {
    zero_f4<<<zeroBlocks, 256, 0, stream>>>((float4*)pool, n4);
    const float* hcur = (rep == 0) ? message : hbuf;
    edge_scatter<<<edgeBlocks, 256, 0, stream>>>(hcur, src, dst, ew, pool, E);
    gemm_gru<<<gemmBlocks, 128, 0, stream>>>(pool, Wn, bn, Wg, bin, brec, hbuf,
                                             out, rep, N);
  }
}